// MemoryUpdate_111669149785
// MI455X (gfx1250) — compile-verified
//
#include <hip/hip_runtime.h>

// Problem dims (fixed by the reference)
#define T_   4
#define B_   32
#define NQ_  1024
#define NKV_ 4
#define D_   512
#define D4_  (D_/4)

typedef __attribute__((ext_vector_type(2))) float v2f;
typedef __attribute__((ext_vector_type(4))) float f4;
typedef __attribute__((ext_vector_type(8))) float v8f;

// ---------------------------------------------------------------------------
// Kernel 1: S[r=n*B+b, d] = (1/T) * sum_t kv[t,b,n,d]   (tiny: 64K elements)
// ---------------------------------------------------------------------------
__global__ void kvsum_kernel(const float* __restrict__ kv, float* __restrict__ S) {
    int idx = blockIdx.x * blockDim.x + threadIdx.x;     // over NKV*B*D = 65536
    if (idx >= NKV_ * B_ * D_) return;
    int d = idx % D_;
    int r = idx / D_;                                    // r = n*B + b
    int n = r / B_;
    int b = r % B_;
    float acc = 0.0f;
#pragma unroll
    for (int t = 0; t < T_; ++t)
        acc += kv[((t * B_ + b) * NKV_ + n) * D_ + d];
    S[idx] = acc * 0.25f;                                // fold the 1/T mean
}

// ---------------------------------------------------------------------------
// Kernel 2: G[r,o] = sum_d S[r,d] * Wg[o,d]  via V_WMMA_F32_16X16X4_F32.
// One wave (32 threads) per 16x16 output tile; grid = (N/16=32, M/16=8).
// A fragment (16x4, MxK):  lanes 0-15 -> K={k0,k0+1}, lanes 16-31 -> K={k0+2,k0+3}
// B fragment (4x16, KxN):  B(k,n) = Wg[n,k]; same half-wave K split, N = lane&15
// ---------------------------------------------------------------------------
__global__ void gemm_wmma_kernel(const float* __restrict__ S,
                                 const float* __restrict__ Wg,
                                 float* __restrict__ G) {
    const int lane = threadIdx.x & 31;
    const int nt   = blockIdx.x;          // 0..31  (N tiles)
    const int mt   = blockIdx.y;          // 0..7   (M tiles)
    const int half = lane >> 4;           // 0 or 1
    const int l15  = lane & 15;
    const int koff = half * 2;            // K sub-offset for this half-wave

    const float* Arow = S  + (mt * 16 + l15) * D_;   // A row m = l15
    const float* Brow = Wg + (nt * 16 + l15) * D_;   // B(k,n)=Wg[n,k] -> row n = l15

    v8f c = {};
#pragma unroll 8
    for (int k0 = 0; k0 < D_; k0 += 4) {
        v2f a = *(const v2f*)(Arow + k0 + koff);     // 8B-aligned (k0+koff even)
        v2f b = *(const v2f*)(Brow + k0 + koff);
        // (neg_a, A, neg_b, B, c_mod, C, reuse_a, reuse_b)
        c = __builtin_amdgcn_wmma_f32_16x16x4_f32(false, a, false, b,
                                                  (short)0, c, false, false);
    }

    // C/D layout: VGPR v holds M=v (lanes 0-15) or M=8+v (lanes 16-31); N = lane&15
    const int col   = nt * 16 + l15;
    const int mbase = mt * 16 + half * 8;
#pragma unroll
    for (int v = 0; v < 8; ++v)
        G[(mbase + v) * D_ + col] = c[v];
}

// ---------------------------------------------------------------------------
// Kernel 3: LIF over T=4.  update[t,b,n,d] = G[t,b,d] * q[t,b,n,d]
//   v = 0.5*(v + update); s = (v >= 0.5); v *= (1-s)   (hard reset)
// float4-vectorized over d; q/out streamed non-temporally (touched once),
// G (256 KB) cached and reused 1024x per (b,d).
// ---------------------------------------------------------------------------
__global__ void lif_kernel(const float* __restrict__ q,
                           const float* __restrict__ G,
                           float* __restrict__ out) {
    const long idx = (long)blockIdx.x * blockDim.x + threadIdx.x; // over B*NQ*D4
    const int  d4  = (int)(idx % D4_);
    const long bn  = idx / D4_;                                   // b*NQ + n
    const int  b   = (int)(bn / NQ_);

    const f4* q4 = (const f4*)q;
    const f4* G4 = (const f4*)G;
    f4*       o4 = (f4*)out;
    const long stride4 = (long)B_ * NQ_ * D4_;                    // per-t stride

    f4 v = {0.0f, 0.0f, 0.0f, 0.0f};
#pragma unroll
    for (int t = 0; t < T_; ++t) {
        f4 g  = G4[((long)(t * B_ + b)) * D4_ + d4];              // cached, reused
        f4 qq = __builtin_nontemporal_load(&q4[(long)t * stride4 + idx]);
        f4 s;
        v.x = 0.5f * (v.x + g.x * qq.x); s.x = (v.x >= 0.5f) ? 1.0f : 0.0f; v.x *= (1.0f - s.x);
        v.y = 0.5f * (v.y + g.y * qq.y); s.y = (v.y >= 0.5f) ? 1.0f : 0.0f; v.y *= (1.0f - s.y);
        v.z = 0.5f * (v.z + g.z * qq.z); s.z = (v.z >= 0.5f) ? 1.0f : 0.0f; v.z *= (1.0f - s.z);
        v.w = 0.5f * (v.w + g.w * qq.w); s.w = (v.w >= 0.5f) ? 1.0f : 0.0f; v.w *= (1.0f - s.w);
        __builtin_nontemporal_store(s, &o4[(long)t * stride4 + idx]);
    }
}

// ---------------------------------------------------------------------------
extern "C" void kernel_launch(void* const* d_in, const int* in_sizes, int n_in,
                              void* d_out, int out_size, void* d_ws, size_t ws_size,
                              hipStream_t stream) {
    const float* q  = (const float*)d_in[0];   // [T,B,NQ,D]
    const float* kv = (const float*)d_in[1];   // [T,B,NKV,D]
    const float* Wg = (const float*)d_in[2];   // [D,D]
    float* out = (float*)d_out;                // [T,B,NQ,D] spikes (fp32 0/1)

    float* S = (float*)d_ws;                   // [NKV*B, D] = 256 KB
    float* G = S + NKV_ * B_ * D_;             // [NKV*B, D] = 256 KB

    // 1) time-sum of kv (with 1/T folded in)
    {
        int n = NKV_ * B_ * D_;
        kvsum_kernel<<<(n + 255) / 256, 256, 0, stream>>>(kv, S);
    }
    // 2) gate GEMM via fp32 WMMA: G = S @ Wg^T
    {
        dim3 grid(D_ / 16, (NKV_ * B_) / 16);  // (32, 8) tiles, 1 wave each
        gemm_wmma_kernel<<<grid, 32, 0, stream>>>(S, Wg, G);
    }
    // 3) streaming LIF (bandwidth-bound: ~512 MB @ 23.3 TB/s ≈ 22 us floor)
    {
        long n4 = (long)B_ * NQ_ * D4_;        // 4,194,304 threads
        lif_kernel<<<(int)(n4 / 256), 256, 0, stream>>>(q, G, out);
    }
}